// ADDsLoss_14628658610644
// MI455X (gfx1250) — compile-verified
//
#include <hip/hip_runtime.h>
#include <hip/hip_bf16.h>
#include <math.h>

// Problem constants (fixed by the reference's setup_inputs()).
#define I_N   64      // B*N instances
#define C_N   8       // classes
#define P_N   1000    // points per model
#define M_N   8       // I_N / C_N  (cap per class)
#define MP    (M_N * P_N)   // 8000 points per class side
#define TILES (MP / 16)     // 500 col tiles of 16
#define TPW   4             // row tiles per wave (500 % 4 == 0)

typedef __attribute__((ext_vector_type(2))) float v2f;
typedef __attribute__((ext_vector_type(8))) float v8f;

#define BIG 1.0e30f
#define EPS2 1.0e-12f

// Bare min instruction: our data is always finite, so skip the compiler's
// sNaN-canonicalization (v_max x,x,x) that fminf/llvm.minnum drags in.
#define VMIN_F32(r, s) asm("v_min_num_f32 %0, %0, %1" : "+v"(r) : "v"(s))

// ---------------------------------------------------------------------------
// Kernel 1: per-class compaction (valid instances first, capped at M),
// n_fg count, and zero the accumulator (harness does not re-zero ws).
// ---------------------------------------------------------------------------
__global__ void adds_compact_kernel(const int* __restrict__ fg,
                                    const int* __restrict__ cls,
                                    int* __restrict__ slot_map,
                                    int* __restrict__ meta,
                                    float* __restrict__ accum) {
    if (blockIdx.x != 0 || threadIdx.x != 0) return;
    int nfg = 0;
    for (int i = 0; i < I_N; ++i) nfg += (fg[i] != 0);
    meta[0] = nfg;
    accum[0] = 0.0f;
    for (int c = 0; c < C_N; ++c) {
        int cnt = 0;
        for (int i = 0; i < I_N && cnt < M_N; ++i)
            if (fg[i] != 0 && cls[i] == c) slot_map[c * M_N + cnt++] = i;
        for (; cnt < M_N; ++cnt) slot_map[c * M_N + cnt] = -1;
    }
}

// ---------------------------------------------------------------------------
// Kernel 2: transform model points by pred/gt poses; store (x,y,z,|q|^2).
// Invalid slots get (0,0,0,BIG): as columns they contribute |b|^2=BIG to the
// minimized quantity (excluded from min); as rows they are skipped later.
// ---------------------------------------------------------------------------
__global__ void adds_transform_kernel(const float* __restrict__ pR,
                                      const float* __restrict__ pT,
                                      const float* __restrict__ gR,
                                      const float* __restrict__ gT,
                                      const float* __restrict__ mpts,
                                      const int* __restrict__ slot_map,
                                      float4* __restrict__ predPts,
                                      float4* __restrict__ gtPts) {
    int idx = blockIdx.x * blockDim.x + threadIdx.x;
    if (idx >= C_N * M_N * P_N) return;
    int p    = idx % P_N;
    int slot = (idx / P_N) % M_N;
    int c    = idx / (P_N * M_N);
    int o    = c * MP + slot * P_N + p;
    int inst = slot_map[c * M_N + slot];
    if (inst < 0) {
        float4 s = make_float4(0.f, 0.f, 0.f, BIG);
        predPts[o] = s;
        gtPts[o]   = s;
        return;
    }
    const float* mp = mpts + (c * P_N + p) * 3;
    float x = mp[0], y = mp[1], z = mp[2];

    const float* R = pR + inst * 9;
    const float* t = pT + inst * 3;
    float qx = R[0]*x + R[1]*y + R[2]*z + t[0];
    float qy = R[3]*x + R[4]*y + R[5]*z + t[1];
    float qz = R[6]*x + R[7]*y + R[8]*z + t[2];
    predPts[o] = make_float4(qx, qy, qz, qx*qx + qy*qy + qz*qz);

    R = gR + inst * 9;
    t = gT + inst * 3;
    qx = R[0]*x + R[1]*y + R[2]*z + t[0];
    qy = R[3]*x + R[4]*y + R[5]*z + t[1];
    qz = R[6]*x + R[7]*y + R[8]*z + t[2];
    gtPts[o] = make_float4(qx, qy, qz, qx*qx + qy*qy + qz*qz);
}

// ---------------------------------------------------------------------------
// Kernel 3/4: one wave handles TPW row tiles (64 rows). WMMA F32 16x16x4 with
// A = [-2x,-2y,-2z,1], B = [x,y,z,|b|^2], C = 0 computes (|b|^2 - 2 a.b).
// Per-lane half selection of (K0,K1) vs (K2,K3) is just a byte offset into
// the stored float4, so the B fragment is a raw b64 load — no packing VALU.
// Running min kept with bare v_min_num_f32 (inline asm): 8 VALU ops per WMMA
// tile, the fundamental floor.  Row norm |a|^2 added back in finalize.
// ---------------------------------------------------------------------------
__global__ void __launch_bounds__(256)
adds_minpass_kernel(const float4* __restrict__ Apts,
                    const float4* __restrict__ Bpts,
                    float* __restrict__ outmin) {
    int gid  = blockIdx.x * blockDim.x + threadIdx.x;
    int w    = gid >> 5;
    int lane = threadIdx.x & 31;
    const int GPC = TILES / TPW;            // 125 wave-groups per class
    if (w >= C_N * GPC) return;             // grid exact; wave-uniform
    int c = w / GPC;
    int g = w % GPC;

    const float4* Ab = Apts + c * MP;
    float*        out = outmin + c * MP;

    bool hi = (lane >= 16);
    // Per-lane base: lo lanes read bytes [0,8) = (x,y); hi lanes [8,16) = (z,w)
    const char* Bbase = (const char*)(Bpts + c * MP) + (hi ? 8 : 0)
                      + (size_t)(lane & 15) * 16;

    // A-matrix 16x4 f32: lanes 0-15 -> (K0,K1)=(-2x,-2y),
    //                    lanes 16-31 -> (K2,K3)=(-2z, 1)
    v2f a[TPW];
    float rmin[TPW][8];
#pragma unroll
    for (int k = 0; k < TPW; ++k) {
        const char* ab = (const char*)(Ab + (g * TPW + k) * 16 + (lane & 15))
                       + (hi ? 8 : 0);
        v2f av = *(const v2f*)ab;
        a[k].x = -2.0f * av.x;
        a[k].y = hi ? 1.0f : (-2.0f * av.y);
#pragma unroll
        for (int v = 0; v < 8; ++v) rmin[k][v] = BIG;
    }

    for (int j = 0; j < TILES; ++j) {
        v2f b = *(const v2f*)(Bbase + (size_t)j * 256);   // 16 pts * 16 B
#pragma unroll
        for (int k = 0; k < TPW; ++k) {
            v8f zero = {};
            v8f acc = __builtin_amdgcn_wmma_f32_16x16x4_f32(
                /*neg_a=*/false, a[k], /*neg_b=*/false, b,
                /*c_mod=*/(short)0, zero, /*reuse_a=*/false, /*reuse_b=*/false);
#pragma unroll
            for (int v = 0; v < 8; ++v) {
                float t = acc[v];
                VMIN_F32(rmin[k][v], t);
            }
        }
    }

    // Cross-lane min within each 16-lane half; VGPR v of tile k holds rows v
    // (lo half) and v+8 (hi half).  Runs once per wave.
#pragma unroll
    for (int k = 0; k < TPW; ++k) {
#pragma unroll
        for (int v = 0; v < 8; ++v) {
            float m = rmin[k][v];
            float s;
            s = __shfl_xor(m, 1, 32);  VMIN_F32(m, s);
            s = __shfl_xor(m, 2, 32);  VMIN_F32(m, s);
            s = __shfl_xor(m, 4, 32);  VMIN_F32(m, s);
            s = __shfl_xor(m, 8, 32);  VMIN_F32(m, s);
            if ((lane & 15) == 0)
                out[(g * TPW + k) * 16 + v + (hi ? 8 : 0)] = m;
        }
    }
}

// ---------------------------------------------------------------------------
// Kernel 5: d^2 = minval + |a_row|^2; per-slot mean of sqrt(max(d2,EPS)),
// 0.5*(mean+mean) accumulated for valid slots.
// ---------------------------------------------------------------------------
__global__ void adds_finalize_kernel(const float* __restrict__ p2g,
                                     const float* __restrict__ g2p,
                                     const float4* __restrict__ predPts,
                                     const float4* __restrict__ gtPts,
                                     const int* __restrict__ slot_map,
                                     float* __restrict__ accum) {
    int slot = blockIdx.x;              // 0 .. C_N*M_N-1
    int c = slot / M_N, s = slot % M_N;
    __shared__ float red[256];
    float sum = 0.0f;
    int base = c * MP + s * P_N;
    for (int i = threadIdx.x; i < P_N; i += blockDim.x) {
        sum += sqrtf(fmaxf(p2g[base + i] + predPts[base + i].w, EPS2));
        sum += sqrtf(fmaxf(g2p[base + i] + gtPts[base + i].w,   EPS2));
    }
    red[threadIdx.x] = sum;
    __syncthreads();
    for (int st = 128; st > 0; st >>= 1) {
        if (threadIdx.x < st) red[threadIdx.x] += red[threadIdx.x + st];
        __syncthreads();
    }
    if (threadIdx.x == 0 && slot_map[slot] >= 0)
        atomicAdd(accum, 0.5f * red[0] / (float)P_N);
}

__global__ void adds_divide_kernel(const float* __restrict__ accum,
                                   const int* __restrict__ meta,
                                   float* __restrict__ out) {
    if (blockIdx.x == 0 && threadIdx.x == 0) {
        int nfg = meta[0];
        out[0] = (nfg > 0) ? (accum[0] / (float)nfg) : 0.0f;
    }
}

// ---------------------------------------------------------------------------
extern "C" void kernel_launch(void* const* d_in, const int* in_sizes, int n_in,
                              void* d_out, int out_size, void* d_ws, size_t ws_size,
                              hipStream_t stream) {
    const float* pR   = (const float*)d_in[0];  // [64,3,3]
    const float* pT   = (const float*)d_in[1];  // [64,3]
    const float* gR   = (const float*)d_in[2];  // [64,3,3]
    const float* gT   = (const float*)d_in[3];  // [64,3]
    const float* mpts = (const float*)d_in[4];  // [8,1000,3]
    const int*   fg   = (const int*)d_in[5];    // [64] bool->int
    const int*   cls  = (const int*)d_in[6];    // [64]
    float* out = (float*)d_out;

    // Workspace layout (all 16-byte aligned; ~2.56 MB total)
    char* ws = (char*)d_ws;
    float4* predPts = (float4*)(ws);                 // 64000 * 16 = 1,024,000
    float4* gtPts   = (float4*)(ws + 1024000);       // 1,024,000
    float*  p2g     = (float*) (ws + 2048000);       // 256,000
    float*  g2p     = (float*) (ws + 2304000);       // 256,000
    int*    slot_map= (int*)   (ws + 2560000);       // 256
    int*    meta    = (int*)   (ws + 2560256);
    float*  accum   = (float*) (ws + 2560512);

    adds_compact_kernel<<<1, 1, 0, stream>>>(fg, cls, slot_map, meta, accum);

    adds_transform_kernel<<<(C_N * M_N * P_N + 255) / 256, 256, 0, stream>>>(
        pR, pT, gR, gT, mpts, slot_map, predPts, gtPts);

    int nwaves = C_N * (TILES / TPW);                // 1000 waves
    int nblk   = (nwaves * 32 + 255) / 256;          // 125 blocks
    adds_minpass_kernel<<<nblk, 256, 0, stream>>>(predPts, gtPts, p2g);
    adds_minpass_kernel<<<nblk, 256, 0, stream>>>(gtPts, predPts, g2p);

    adds_finalize_kernel<<<C_N * M_N, 256, 0, stream>>>(
        p2g, g2p, predPts, gtPts, slot_map, accum);
    adds_divide_kernel<<<1, 1, 0, stream>>>(accum, meta, out);
}